// HMLoss_1778116461180
// MI455X (gfx1250) — compile-verified
//
#include <hip/hip_runtime.h>

#define HH 2048
#define NPIX (HH * HH)
#define NB 256

typedef __attribute__((ext_vector_type(2))) float v2f;
typedef __attribute__((ext_vector_type(8))) float v8f;

// ---------- helpers ----------
__device__ __forceinline__ float dn255(float x) {
    // (x+1)/2 clamped to [0,1], *255
    float t = (x + 1.0f) * 0.5f;
    t = fminf(fmaxf(t, 0.0f), 1.0f);
    return t * 255.0f;
}

__device__ __forceinline__ int bin256(float v) {
    // np.histogram(bins=256, range=(0,255)): bin = floor(v*256/255), clipped
    int b = (int)floorf(v * (256.0f / 255.0f));
    b = b < 0 ? 0 : b;
    return b > (NB - 1) ? (NB - 1) : b;
}

// Region id for mask_A labels (disjoint label sets): 0=face,1=hair,2=eyeL,3=eyeR,-1=none
__device__ __forceinline__ int regionA(int l) {
    const unsigned FACE = (1u << 1) | (1u << 7) | (1u << 8) | (1u << 10) | (1u << 11) | (1u << 14);
    const unsigned HAIR = (1u << 17);
    const unsigned EYEL = (1u << 2) | (1u << 4);
    const unsigned EYER = (1u << 3) | (1u << 5);
    unsigned b = 1u << (l & 31);
    if (b & FACE) return 0;
    if (b & HAIR) return 1;
    if (b & EYEL) return 2;
    if (b & EYER) return 3;
    return -1;
}

// mask_B region weights (face keeps the reference's (mask_A==11) quirk; can be 2)
__device__ __forceinline__ void maskB(int la, int lb, int mv[4]) {
    const unsigned FACEB = (1u << 1) | (1u << 7) | (1u << 8) | (1u << 10) | (1u << 14);
    const unsigned HAIR = (1u << 17);
    const unsigned EYEL = (1u << 2) | (1u << 4);
    const unsigned EYER = (1u << 3) | (1u << 5);
    unsigned b = 1u << (lb & 31);
    mv[0] = ((b & FACEB) ? 1 : 0) + ((la == 11) ? 1 : 0);
    mv[1] = (b & HAIR) ? 1 : 0;
    mv[2] = (b & EYEL) ? 1 : 0;
    mv[3] = (b & EYER) ? 1 : 0;
}

__device__ __forceinline__ void f4arr(float4 v, float* o) { o[0] = v.x; o[1] = v.y; o[2] = v.z; o[3] = v.w; }

// ---------- kernels ----------
__global__ void init_ws_kernel(unsigned* __restrict__ ws, int n) {
    int i = blockIdx.x * blockDim.x + threadIdx.x;
    if (i < n) ws[i] = 0u;
}

// Fused histogram pass: 24 histograms (4 regions x 3 channels x {src,tar}) in LDS,
// flushed (nonzero bins only) to global with atomics.
__global__ void __launch_bounds__(256) hist_kernel(const float* __restrict__ fa,
                                                   const float* __restrict__ rb,
                                                   const int* __restrict__ ma,
                                                   const int* __restrict__ mb,
                                                   unsigned* __restrict__ ghist /* [24*256]: src(12*256) then tar(12*256) */) {
    __shared__ unsigned h[24 * NB];
    const int tid = threadIdx.x;
    for (int i = tid; i < 24 * NB; i += 256) h[i] = 0u;
    __syncthreads();

    const int nvec = NPIX / 4;
    const int stride = gridDim.x * 256;
    for (int p4 = blockIdx.x * 256 + tid; p4 < nvec; p4 += stride) {
        // prefetch next iteration's lines (global_prefetch_b8)
        if (p4 + stride < nvec) {
            __builtin_prefetch(&((const float4*)fa)[p4 + stride], 0, 0);
            __builtin_prefetch(&((const float4*)rb)[p4 + stride], 0, 0);
        }
        int4 la4 = ((const int4*)ma)[p4];
        int4 lb4 = ((const int4*)mb)[p4];
        float vA[3][4], vB[3][4];
        f4arr(((const float4*)(fa + 0 * NPIX))[p4], vA[0]);
        f4arr(((const float4*)(fa + 1 * NPIX))[p4], vA[1]);
        f4arr(((const float4*)(fa + 2 * NPIX))[p4], vA[2]);
        f4arr(((const float4*)(rb + 0 * NPIX))[p4], vB[0]);
        f4arr(((const float4*)(rb + 1 * NPIX))[p4], vB[1]);
        f4arr(((const float4*)(rb + 2 * NPIX))[p4], vB[2]);
        int la[4] = {la4.x, la4.y, la4.z, la4.w};
        int lb[4] = {lb4.x, lb4.y, lb4.z, lb4.w};

#pragma unroll
        for (int k = 0; k < 4; ++k) {
            // source-side (mask_A regions, weight 0/1, disjoint)
            int r = regionA(la[k]);
            if (r >= 0) {
#pragma unroll
                for (int c = 0; c < 3; ++c) {
                    int b = bin256(dn255(vA[c][k]));
                    atomicAdd(&h[(r * 3 + c) * NB + b], 1u);
                }
            }
            // target-side (mask_B regions; face weight can be 2 and overlap others)
            int mv[4];
            maskB(la[k], lb[k], mv);
#pragma unroll
            for (int rr = 0; rr < 4; ++rr) {
                if (mv[rr] > 0) {
                    float w = (float)mv[rr];
#pragma unroll
                    for (int c = 0; c < 3; ++c) {
                        int b = bin256(dn255(vB[c][k]) * w);
                        atomicAdd(&h[12 * NB + (rr * 3 + c) * NB + b], 1u);
                    }
                }
            }
        }
    }
    __syncthreads();
    for (int i = tid; i < 24 * NB; i += 256) {
        unsigned v = h[i];
        if (v) atomicAdd(&ghist[i], v);
    }
}

// One block per (region, channel): CDFs + transfer table search.
__global__ void __launch_bounds__(256) table_kernel(const unsigned* __restrict__ ghist,
                                                    int* __restrict__ tables) {
    const int rc = blockIdx.x;  // 0..11
    const int i = threadIdx.x;  // 0..255
    __shared__ float d[NB], a[NB];
    d[i] = (float)ghist[rc * NB + i];
    a[i] = (float)ghist[12 * NB + rc * NB + i];
    __syncthreads();
    if (i == 0) {
        float s = 0.f;
        for (int k = 0; k < NB; ++k) s += d[k];
        s = fmaxf(s, 1.0f);
        float cum = 0.f;
        for (int k = 0; k < NB; ++k) { cum += d[k] / s; d[k] = cum; }
    }
    if (i == 1) {
        float s = 0.f;
        for (int k = 0; k < NB; ++k) s += a[k];
        s = fmaxf(s, 1.0f);
        float cum = 0.f;
        for (int k = 0; k < NB; ++k) { cum += a[k] / s; a[k] = cum; }
    }
    __syncthreads();
    float di = d[i];
    int t = i;
    bool found = false;
    for (int j = 1; j < NB; ++j) {
        if (!found && di >= a[j - 1] && di <= a[j]) { t = j; found = true; }
    }
    if (i == 0) t = 0;
    if (i == NB - 1) t = NB - 1;
    tables[rc * NB + i] = t;
}

// Fused lookup + |diff| accumulation.  Block reduction uses V_WMMA_F32_16X16X4_F32
// (A = 64 partials, B = ones -> exact f32 row sums), 4 WMMAs cover 256 partials.
__global__ void __launch_bounds__(256) loss_kernel(const float* __restrict__ fa,
                                                   const int* __restrict__ ma,
                                                   const int* __restrict__ tables,
                                                   float* __restrict__ acc) {
    const int tid = threadIdx.x;
    float local = 0.f;
    const int nvec = NPIX / 4;
    const int stride = gridDim.x * 256;
    for (int p4 = blockIdx.x * 256 + tid; p4 < nvec; p4 += stride) {
        if (p4 + stride < nvec) __builtin_prefetch(&((const float4*)fa)[p4 + stride], 0, 0);
        int4 la4 = ((const int4*)ma)[p4];
        float vA[3][4];
        f4arr(((const float4*)(fa + 0 * NPIX))[p4], vA[0]);
        f4arr(((const float4*)(fa + 1 * NPIX))[p4], vA[1]);
        f4arr(((const float4*)(fa + 2 * NPIX))[p4], vA[2]);
        int la[4] = {la4.x, la4.y, la4.z, la4.w};
#pragma unroll
        for (int k = 0; k < 4; ++k) {
            int r = regionA(la[k]);
            if (r >= 0) {
#pragma unroll
                for (int c = 0; c < 3; ++c) {
                    float v = dn255(vA[c][k]);
                    int idx = (int)v;               // v >= 0, trunc == floor
                    idx = idx > (NB - 1) ? (NB - 1) : idx;
                    float looked = (float)tables[(r * 3 + c) * NB + idx];
                    local += fabsf(v - looked);
                }
            }
        }
    }

    __shared__ float red[256];
    red[tid] = local;
    __syncthreads();

    if (tid < 32) {  // wave 0, EXEC all ones within the wave (WMMA requirement)
#if __has_builtin(__builtin_amdgcn_wmma_f32_16x16x4_f32)
        v8f cacc = {0.f, 0.f, 0.f, 0.f, 0.f, 0.f, 0.f, 0.f};
#pragma unroll
        for (int k = 0; k < 4; ++k) {
            v2f av; av[0] = red[k * 64 + tid * 2]; av[1] = red[k * 64 + tid * 2 + 1];
            v2f bv; bv[0] = 1.0f; bv[1] = 1.0f;
            // D[m][n] = rowsum(m) when B == ones; exact f32 accumulation through C
            cacc = __builtin_amdgcn_wmma_f32_16x16x4_f32(
                false, av, false, bv, (short)0, cacc, false, false);
        }
        // lane n in 0..15 holds M=0..7 in c[0..7]; lane n+16 holds M=8..15
        float s = cacc[0] + cacc[1] + cacc[2] + cacc[3] + cacc[4] + cacc[5] + cacc[6] + cacc[7];
        red[tid] = s;
#else
        float s = 0.f;
        for (int k = tid; k < 256; k += 32) s += red[k];
        red[tid] = s;
#endif
    }
    __syncthreads();
    if (tid == 0) {
#if __has_builtin(__builtin_amdgcn_wmma_f32_16x16x4_f32)
        atomicAdd(acc, red[0] + red[16]);
#else
        float s = 0.f;
        for (int k = 0; k < 32; ++k) s += red[k];
        atomicAdd(acc, s);
#endif
    }
}

__global__ void final_kernel(const float* __restrict__ acc, float* __restrict__ out) {
    // loss = 0.1*(face + hair + eyeL + eyeR), each mean over 3*H*W
    out[0] = 0.1f * acc[0] * (1.0f / (3.0f * (float)NPIX));
}

// ---------- launcher ----------
extern "C" void kernel_launch(void* const* d_in, const int* in_sizes, int n_in,
                              void* d_out, int out_size, void* d_ws, size_t ws_size,
                              hipStream_t stream) {
    (void)in_sizes; (void)n_in; (void)out_size; (void)ws_size;
    const float* fake_A = (const float*)d_in[0];
    const float* ref_B  = (const float*)d_in[1];
    const int*   mask_A = (const int*)d_in[2];
    const int*   mask_B = (const int*)d_in[3];

    // workspace layout (32-bit words):
    //   [0, 24*256)            : histograms (src 12*256, then tar 12*256)
    //   [24*256, 36*256)       : transfer tables (int)
    //   [36*256]               : float accumulator
    unsigned* ghist  = (unsigned*)d_ws;
    int*      tables = (int*)(ghist + 24 * NB);
    float*    acc    = (float*)(tables + 12 * NB);

    const int n_words = 24 * NB + 12 * NB + 1;
    init_ws_kernel<<<(n_words + 255) / 256, 256, 0, stream>>>((unsigned*)d_ws, n_words);

    hist_kernel<<<1024, 256, 0, stream>>>(fake_A, ref_B, mask_A, mask_B, ghist);
    table_kernel<<<12, 256, 0, stream>>>(ghist, tables);
    loss_kernel<<<2048, 256, 0, stream>>>(fake_A, mask_A, tables, acc);
    final_kernel<<<1, 1, 0, stream>>>(acc, (float*)d_out);
}